// AlphaNet_71614284693954
// MI455X (gfx1250) — compile-verified
//
#include <hip/hip_runtime.h>
#include <hip/hip_bf16.h>

// ---------------------------------------------------------------------------
// Sizes from the reference
// ---------------------------------------------------------------------------
#define BB   65536
#define FF   8
#define TT   60
#define DD   10
#define WW   6
#define NCH  76          // 28 corr + 8 xb4 + 8 xb6 + 32 xb9
#define HRAW (NCH*WW)    // 456 floats per batch
#define KDIM 608         // 456 + 152
#define EPSC 1e-8f
#define BNEPS 1e-5f

typedef __attribute__((ext_vector_type(2))) float v2f;
typedef __attribute__((ext_vector_type(8))) float v8f;

// triu_indices(8, k=1) row-major
__device__ __constant__ unsigned char PI_[28] =
  {0,0,0,0,0,0,0, 1,1,1,1,1,1, 2,2,2,2,2, 3,3,3,3, 4,4,4, 5,5, 6};
__device__ __constant__ unsigned char PJ_[28] =
  {1,2,3,4,5,6,7, 2,3,4,5,6,7, 3,4,5,6,7, 4,5,6,7, 5,6,7, 6,7, 7};

// ---------------------------------------------------------------------------
// K1: per-batch raw features.  One wave32 per batch, 4 waves / block.
// ---------------------------------------------------------------------------
__global__ void __launch_bounds__(128) k1_features(
    const float* __restrict__ xb, const float* __restrict__ conv_w,
    const float* __restrict__ conv_b, float* __restrict__ h_raw)
{
    __shared__ float sx[4][480];
    __shared__ float smu[4][48];
    __shared__ float sstd[4][48];
    __shared__ float scw[40];
    __shared__ float scb[4];

    const int wv   = threadIdx.x >> 5;
    const int lane = threadIdx.x & 31;
    const int b    = blockIdx.x * 4 + wv;

    if (threadIdx.x < 40) scw[threadIdx.x] = conv_w[threadIdx.x];
    if (threadIdx.x < 4)  scb[threadIdx.x] = conv_b[threadIdx.x];

    const float* xsrc = xb + (size_t)b * 480;
    #pragma unroll
    for (int i = 0; i < 15; ++i) sx[wv][lane + 32*i] = xsrc[lane + 32*i];
    __syncthreads();

    float* hb = h_raw + (size_t)b * HRAW;

    // pass A: per (f,w) mean / std / xb4 / xb6
    for (int u = lane; u < 48; u += 32) {
        const int f = u / 6, w = u % 6;
        const float* xp = &sx[wv][f*60 + w*10];
        float s = 0.f;
        #pragma unroll
        for (int d = 0; d < DD; ++d) s += xp[d];
        const float mu = s * 0.1f;
        float ss = 0.f, acc = 0.f;
        #pragma unroll
        for (int d = 0; d < DD; ++d) {
            const float c = xp[d] - mu;
            ss  += c * c;
            acc += xp[d] * ((float)(d + 1) * (1.0f / 55.0f));
        }
        const float sd = sqrtf(ss * 0.1f + EPSC);
        smu[wv][u]  = mu;
        sstd[wv][u] = sd;
        hb[(28 + f) * 6 + w] = mu / (sd + EPSC);   // xb4
        hb[(36 + f) * 6 + w] = acc;                // xb6
    }
    __syncthreads();

    // pass B: 28 pairwise correlations x 6 windows
    for (int v = lane; v < 168; v += 32) {
        const int p = v / 6, w = v % 6;
        const int i = PI_[p], j = PJ_[p];
        const float* xi = &sx[wv][i*60 + w*10];
        const float* xj = &sx[wv][j*60 + w*10];
        const float mi = smu[wv][i*6 + w], mj = smu[wv][j*6 + w];
        float cov = 0.f;
        #pragma unroll
        for (int d = 0; d < DD; ++d) cov += (xi[d] - mi) * (xj[d] - mj);
        cov *= 0.1f;
        hb[p * 6 + w] = cov / (sstd[wv][i*6+w] * sstd[wv][j*6+w] + EPSC);
    }

    // pass C: 1xD conv -> 32 channels x 6 windows
    for (int v = lane; v < 192; v += 32) {
        const int o = v / 48, rem = v % 48, f = rem / 6, w = rem % 6;
        const float* xp = &sx[wv][f*60 + w*10];
        float acc = scb[o];
        #pragma unroll
        for (int d = 0; d < DD; ++d) acc += xp[d] * scw[o*10 + d];
        hb[(44 + o*8 + f) * 6 + w] = acc;
    }
}

// ---------------------------------------------------------------------------
// K2a: per-channel partial (sum, sumsq) of h over (B,W).  grid (76, 64)
// ---------------------------------------------------------------------------
__global__ void __launch_bounds__(256) k2a_hstats(
    const float* __restrict__ h_raw, float* __restrict__ partial1)
{
    const int c = blockIdx.x, slice = blockIdx.y, tid = threadIdx.x;
    float s = 0.f, q = 0.f;
    for (int b = slice * 1024 + tid; b < (slice + 1) * 1024; b += 256) {
        const float* p = h_raw + (size_t)b * HRAW + c * 6;
        #pragma unroll
        for (int w = 0; w < 6; ++w) { const float v = p[w]; s += v; q += v*v; }
    }
    __shared__ float rs[256], rq[256];
    rs[tid] = s; rq[tid] = q; __syncthreads();
    for (int st = 128; st; st >>= 1) {
        if (tid < st) { rs[tid] += rs[tid + st]; rq[tid] += rq[tid + st]; }
        __syncthreads();
    }
    if (tid == 0) {
        partial1[(c * 64 + slice) * 2 + 0] = rs[0];
        partial1[(c * 64 + slice) * 2 + 1] = rq[0];
    }
}

// K2b: finalize BN params for h.  grid 76, block 64.
__global__ void __launch_bounds__(64) k2b_hfinal(
    const float* __restrict__ partial1,
    const float* __restrict__ bn1_g, const float* __restrict__ bn1_b,
    const float* __restrict__ bn4_g, const float* __restrict__ bn4_b,
    const float* __restrict__ bn6_g, const float* __restrict__ bn6_b,
    const float* __restrict__ bn9_g, const float* __restrict__ bn9_b,
    float* __restrict__ ss1)
{
    const int c = blockIdx.x, tid = threadIdx.x;
    __shared__ float rs[64], rq[64];
    rs[tid] = partial1[(c * 64 + tid) * 2 + 0];
    rq[tid] = partial1[(c * 64 + tid) * 2 + 1];
    __syncthreads();
    for (int st = 32; st; st >>= 1) {
        if (tid < st) { rs[tid] += rs[tid + st]; rq[tid] += rq[tid + st]; }
        __syncthreads();
    }
    if (tid == 0) {
        const float N = (float)BB * 6.0f;
        const float mean = rs[0] / N;
        const float var  = rq[0] / N - mean * mean;
        float g, bb;
        if      (c < 28) { g = bn1_g[c];      bb = bn1_b[c];      }
        else if (c < 36) { g = bn4_g[c - 28]; bb = bn4_b[c - 28]; }
        else if (c < 44) { g = bn6_g[c - 36]; bb = bn6_b[c - 36]; }
        else             { g = bn9_g[c - 44]; bb = bn9_b[c - 44]; }
        const float scale = g * rsqrtf(var + BNEPS);
        ss1[c * 2 + 0] = scale;
        ss1[c * 2 + 1] = bb - mean * scale;
    }
}

// ---------------------------------------------------------------------------
// K3a: BN(h) -> max over window triples -> partial stats.  grid (76, 128)
// ---------------------------------------------------------------------------
__global__ void __launch_bounds__(256) k3a_mstats(
    const float* __restrict__ h_raw, const float* __restrict__ ss1,
    float* __restrict__ partial2)
{
    const int c = blockIdx.x, slice = blockIdx.y, tid = threadIdx.x;
    const float sc = ss1[c * 2], sh = ss1[c * 2 + 1];
    float s = 0.f, q = 0.f;
    for (int b = slice * 512 + tid; b < (slice + 1) * 512; b += 256) {
        const float* p = h_raw + (size_t)b * HRAW + c * 6;
        float y[6];
        #pragma unroll
        for (int w = 0; w < 6; ++w) y[w] = p[w] * sc + sh;
        const float m0 = fmaxf(y[0], fmaxf(y[1], y[2]));
        const float m1 = fmaxf(y[3], fmaxf(y[4], y[5]));
        s += m0 + m1; q += m0*m0 + m1*m1;
    }
    __shared__ float rs[256], rq[256];
    rs[tid] = s; rq[tid] = q; __syncthreads();
    for (int st = 128; st; st >>= 1) {
        if (tid < st) { rs[tid] += rs[tid + st]; rq[tid] += rq[tid + st]; }
        __syncthreads();
    }
    if (tid == 0) {
        partial2[(c * 128 + slice) * 2 + 0] = rs[0];
        partial2[(c * 128 + slice) * 2 + 1] = rq[0];
    }
}

// K3b: finalize BN params for h_max.  grid 76, block 128.
__global__ void __launch_bounds__(128) k3b_mfinal(
    const float* __restrict__ partial2,
    const float* __restrict__ bnmax_g, const float* __restrict__ bnmax_b,
    float* __restrict__ ss2)
{
    const int c = blockIdx.x, tid = threadIdx.x;
    __shared__ float rs[128], rq[128];
    rs[tid] = partial2[(c * 128 + tid) * 2 + 0];
    rq[tid] = partial2[(c * 128 + tid) * 2 + 1];
    __syncthreads();
    for (int st = 64; st; st >>= 1) {
        if (tid < st) { rs[tid] += rs[tid + st]; rq[tid] += rq[tid + st]; }
        __syncthreads();
    }
    if (tid == 0) {
        const float N = (float)BB * 2.0f;
        const float mean = rs[0] / N;
        const float var  = rq[0] / N - mean * mean;
        const float scale = bnmax_g[c] * rsqrtf(var + BNEPS);
        ss2[c * 2 + 0] = scale;
        ss2[c * 2 + 1] = bnmax_b[c] - mean * scale;
    }
}

// ---------------------------------------------------------------------------
// K5: fused BN + flat assembly + WMMA GEMM (608 -> 30) + ReLU + (30 -> 1).
// One wave32 per 16-batch tile; 2 waves / block.
// LDS layout (floats):
//   w1s : 304*32*2 = 19456   (float2 [k/2][n]:  (w1[n][2k2], w1[n][2k2+1]))
//   b1s : 32, w2s : 32       (zero padded to N=32)
//   fbuf: per-wave 304*16*2  (float2 [k/2][m] of the flat vector)
// ---------------------------------------------------------------------------
#define W1S_FLOATS (304*32*2)
#define FB_FLOATS  (304*16*2)

__global__ void __launch_bounds__(64) k5_mlp(
    const float* __restrict__ h_raw, const float* __restrict__ ss1,
    const float* __restrict__ ss2,  const float* __restrict__ w1,
    const float* __restrict__ b1,   const float* __restrict__ w2,
    const float* __restrict__ b2,   float* __restrict__ out)
{
    extern __shared__ float smem[];
    float* w1s  = smem;                       // 19456
    float* b1s  = w1s + W1S_FLOATS;           // 32
    float* w2s  = b1s + 32;                   // 32
    float* fbse = w2s + 32;                   // 2 * FB_FLOATS

    const int wv = threadIdx.x >> 5, lane = threadIdx.x & 31;
    float* fbuf = fbse + wv * FB_FLOATS;

    // Stage w1 in WMMA B layout (B[k][n] = w1[n][k]); pad n to 32 with zeros.
    for (int idx = threadIdx.x; idx < 32 * KDIM; idx += 64) {
        const int n = idx / KDIM, k = idx % KDIM;
        const float v = (n < 30) ? w1[n * KDIM + k] : 0.0f;
        w1s[(k >> 1) * 64 + n * 2 + (k & 1)] = v;
    }
    if (threadIdx.x < 32) {
        b1s[threadIdx.x] = (threadIdx.x < 30) ? b1[threadIdx.x] : 0.0f;
        w2s[threadIdx.x] = (threadIdx.x < 30) ? w2[threadIdx.x] : 0.0f;
    }
    __syncthreads();

    const int tile = blockIdx.x * 2 + wv;
    const int m0   = tile * 16;

    // flat[0..455] = BN(h_raw), stored as float2 [k/2][m]
    for (int m = 0; m < 16; ++m) {
        const float* hp = h_raw + (size_t)(m0 + m) * HRAW;
        for (int idx = lane; idx < HRAW; idx += 32) {
            const int c = idx / 6;
            const float v = hp[idx] * ss1[c * 2] + ss1[c * 2 + 1];
            fbuf[(idx >> 1) * 32 + m * 2 + (idx & 1)] = v;
        }
    }
    __syncthreads();

    // flat[456..607] = BN(max over triples of BN(h))
    for (int idx = lane; idx < 152; idx += 32) {
        const int c = idx >> 1, half = idx & 1;
        const int kb = c * 6 + half * 3;
        const float sc = ss2[c * 2], sh = ss2[c * 2 + 1];
        for (int m = 0; m < 16; ++m) {
            const float y0 = fbuf[((kb    ) >> 1) * 32 + m * 2 + ((kb    ) & 1)];
            const float y1 = fbuf[((kb + 1) >> 1) * 32 + m * 2 + ((kb + 1) & 1)];
            const float y2 = fbuf[((kb + 2) >> 1) * 32 + m * 2 + ((kb + 2) & 1)];
            const float mx = fmaxf(y0, fmaxf(y1, y2));
            const int k = 456 + idx;
            fbuf[(k >> 1) * 32 + m * 2 + (k & 1)] = mx * sc + sh;
        }
    }
    __syncthreads();

    // GEMM: D(16x32) = A(16x608) x B(608x32) via V_WMMA_F32_16X16X4_F32
    v8f acc0 = {};
    v8f acc1 = {};
    const v2f* fa = (const v2f*)fbuf;   // [k2 * 16 + m]
    const v2f* fb = (const v2f*)w1s;    // [k2 * 32 + n]
    const int g = lane >> 4, q = lane & 15;
    for (int k2 = 0; k2 < 304; k2 += 2) {       // 4 K-values per step
        const v2f a  = fa[(k2 + g) * 16 + q];
        const v2f b0 = fb[(k2 + g) * 32 + q];
        const v2f b1v = fb[(k2 + g) * 32 + q + 16];
        acc0 = __builtin_amdgcn_wmma_f32_16x16x4_f32(
            false, a, false, b0,  (short)0, acc0, false, false);
        acc1 = __builtin_amdgcn_wmma_f32_16x16x4_f32(
            false, a, false, b1v, (short)0, acc1, false, false);
    }

    // Epilogue: relu(hidden + b1) . w2 + b2, reduced across the 16-lane groups
    const float bias2 = b2[0];
    const float w2a = w2s[q], w2b = w2s[q + 16];
    const float b1a = b1s[q], b1b = b1s[q + 16];
    #pragma unroll
    for (int r = 0; r < 8; ++r) {
        float v = fmaxf(acc0[r] + b1a, 0.0f) * w2a
                + fmaxf(acc1[r] + b1b, 0.0f) * w2b;
        #pragma unroll
        for (int s = 1; s < 16; s <<= 1) v += __shfl_xor(v, s, 32);
        if (q == 0) out[m0 + r + 8 * g] = v + bias2;
    }
}

// ---------------------------------------------------------------------------
// Host-side launcher
// ---------------------------------------------------------------------------
extern "C" void kernel_launch(void* const* d_in, const int* in_sizes, int n_in,
                              void* d_out, int out_size, void* d_ws, size_t ws_size,
                              hipStream_t stream) {
    const float* xb      = (const float*)d_in[0];
    const float* conv_w  = (const float*)d_in[1];
    const float* conv_b  = (const float*)d_in[2];
    const float* bn1_g   = (const float*)d_in[3];
    const float* bn1_b   = (const float*)d_in[4];
    const float* bn4_g   = (const float*)d_in[5];
    const float* bn4_b   = (const float*)d_in[6];
    const float* bn6_g   = (const float*)d_in[7];
    const float* bn6_b   = (const float*)d_in[8];
    const float* bn9_g   = (const float*)d_in[9];
    const float* bn9_b   = (const float*)d_in[10];
    const float* bnmax_g = (const float*)d_in[11];
    const float* bnmax_b = (const float*)d_in[12];
    const float* w1      = (const float*)d_in[13];
    const float* b1      = (const float*)d_in[14];
    const float* w2      = (const float*)d_in[15];
    const float* b2      = (const float*)d_in[16];
    float* out = (float*)d_out;

    float* ws       = (float*)d_ws;
    float* h_raw    = ws;                                  // B*456
    float* partial1 = h_raw + (size_t)BB * HRAW;           // 76*64*2
    float* ss1      = partial1 + 76 * 64 * 2;              // 76*2
    float* partial2 = ss1 + 76 * 2;                        // 76*128*2
    float* ss2      = partial2 + 76 * 128 * 2;             // 76*2

    k1_features<<<BB / 4, 128, 0, stream>>>(xb, conv_w, conv_b, h_raw);
    k2a_hstats<<<dim3(76, 64), 256, 0, stream>>>(h_raw, partial1);
    k2b_hfinal<<<76, 64, 0, stream>>>(partial1, bn1_g, bn1_b, bn4_g, bn4_b,
                                      bn6_g, bn6_b, bn9_g, bn9_b, ss1);
    k3a_mstats<<<dim3(76, 128), 256, 0, stream>>>(h_raw, ss1, partial2);
    k3b_mfinal<<<76, 128, 0, stream>>>(partial2, bnmax_g, bnmax_b, ss2);

    const size_t smem = (size_t)(W1S_FLOATS + 64 + 2 * FB_FLOATS) * sizeof(float);
    k5_mlp<<<BB / 32, 64, smem, stream>>>(h_raw, ss1, ss2, w1, b1, w2, b2, out);
}